// LGMDNet_60885456388151
// MI455X (gfx1250) — compile-verified
//
#include <hip/hip_runtime.h>
#include <hip/hip_bf16.h>

typedef __attribute__((ext_vector_type(2))) float v2f;
typedef __attribute__((ext_vector_type(8))) float v8f;

#define H_IN   260
#define W_IN   346
#define H_OUT  65
#define W_OUT  86
#define NPIX   (H_OUT * W_OUT)   // 5590
#define T_DIM  100
#define B_DIM  16

// ---------------------------------------------------------------------------
// Stage 1: fused AvgPool(4) + 3x3 exc conv (2ch -> 1ch, SAME) + ReLU.
// 32x32 output tile per block; pooled patch (+1 halo) built in LDS so each
// element of x is read ~once (halo overhead ~13%). Carries nearly all HBM
// traffic (~1.3 GB read at 23.3 TB/s -> this stage is the runtime).
// ---------------------------------------------------------------------------
__global__ __launch_bounds__(256) void pool_exc_kernel(
    const float* __restrict__ x, const float* __restrict__ exc_w,
    float* __restrict__ exc) {
  const int tb  = blockIdx.z;                 // t*B + b
  const int ty0 = blockIdx.y * 32;
  const int tx0 = blockIdx.x * 32;

  __shared__ float pooled[2][34][34];

  const size_t imgBase = (size_t)tb * 2 * H_IN * W_IN;

  for (int i = threadIdx.x; i < 2 * 34 * 34; i += 256) {
    const int ch  = i / (34 * 34);
    const int rem = i % (34 * 34);
    const int py  = rem / 34;
    const int px  = rem % 34;
    const int gpy = ty0 - 1 + py;
    const int gpx = tx0 - 1 + px;
    float s = 0.0f;
    if (gpy >= 0 && gpy < H_OUT && gpx >= 0 && gpx < W_OUT) {
      const float* base = x + imgBase + (size_t)ch * (H_IN * W_IN)
                            + (size_t)(gpy * 4) * W_IN + (gpx * 4);
#pragma unroll
      for (int rr = 0; rr < 4; ++rr) {
        const float* row = base + rr * W_IN;
        s += row[0] + row[1] + row[2] + row[3];
      }
      s *= 0.0625f;
    }
    pooled[ch][py][px] = s;
  }
  __syncthreads();

  float w0[9], w1[9];
#pragma unroll
  for (int k = 0; k < 9; ++k) { w0[k] = exc_w[k]; w1[k] = exc_w[9 + k]; }

  float* out = exc + (size_t)tb * NPIX;
  for (int i = threadIdx.x; i < 32 * 32; i += 256) {
    const int ly = i / 32, lx = i % 32;
    const int gy = ty0 + ly, gx = tx0 + lx;
    if (gy < H_OUT && gx < W_OUT) {
      float acc = 0.0f;
#pragma unroll
      for (int dy = 0; dy < 3; ++dy)
#pragma unroll
        for (int dx = 0; dx < 3; ++dx)
          acc += w0[dy * 3 + dx] * pooled[0][ly + dy][lx + dx] +
                 w1[dy * 3 + dx] * pooled[1][ly + dy][lx + dx];
      out[gy * W_OUT + gx] = fmaxf(acc, 0.0f);
    }
  }
}

// ---------------------------------------------------------------------------
// Stage 2: lgmd_in[t] = exc[t] + conv7x7(exc[t-1], inh_w)  (SAME, pad 3).
// Toeplitz-GEMM via V_WMMA_F32_16X16X4_F32:
//   out[m,n] = sum_dy sum_j patch[m+dy][j] * Bmat[dy][n][j]
// Bmat (banded Toeplitz, Bmat[dy][n][j] = inh_w[dy][j-n] for 0<=j-n<=6, else
// 0) is tile-independent and expanded ONCE into LDS so the hot loop has zero
// divergence: per WMMA just two aligned ds_load_b64 (A pair, B pair) and the
// matrix op. 7 dy x 6 K-chunks = 42 WMMAs per 16x16 tile, one wave per tile,
// EXEC all-ones (uniform branches only).
// ---------------------------------------------------------------------------
__global__ __launch_bounds__(32) void lgmd_inh_wmma_kernel(
    const float* __restrict__ exc, const float* __restrict__ inh_w,
    float* __restrict__ lgmd) {
  const int tb   = blockIdx.z;            // t*B + b
  const int t    = tb / B_DIM;
  const int x0   = blockIdx.x * 16;
  const int y0   = blockIdx.y * 16;
  const int lane = threadIdx.x;

  __shared__ float patch[22][24];         // exc[t-1] slab, zero padded
  __shared__ float wsm[49];               // raw 7x7 weights
  __shared__ float Bmat[7][16][24];       // expanded Toeplitz [dy][n][j]

  const float* excT = exc  + (size_t)tb * NPIX;
  float*       out  = lgmd + (size_t)tb * NPIX;

  v8f acc = {};  // C/D accumulator (16x16 f32)

  if (t > 0) {
    const float* excP = exc + (size_t)(tb - B_DIM) * NPIX;  // exc[t-1]

    // -- patch load (22 valid cols, cols 22..23 are K padding = 0)
    for (int i = lane; i < 22 * 24; i += 32) {
      const int r  = i / 24;
      const int cc = i % 24;
      const int gy = y0 - 3 + r;
      const int gx = x0 - 3 + cc;
      float v = 0.0f;
      if (cc < 22 && gy >= 0 && gy < H_OUT && gx >= 0 && gx < W_OUT)
        v = excP[gy * W_OUT + gx];
      patch[r][cc] = v;
    }
    for (int i = lane; i < 49; i += 32) wsm[i] = inh_w[i];
    __syncthreads();

    // -- expand banded Toeplitz into LDS (uniform, unconditional hot loop)
    for (int i = lane; i < 7 * 16 * 24; i += 32) {
      const int dy  = i / (16 * 24);
      const int rem = i % (16 * 24);
      const int n   = rem / 24;
      const int j   = rem % 24;
      const int d   = j - n;
      Bmat[dy][n][j] = (d >= 0 && d < 7) ? wsm[dy * 7 + d] : 0.0f;
    }
    __syncthreads();

    const int m  = lane & 15;   // A row (lanes 0-15) / repeated (16-31)
    const int hi = lane >> 4;   // lane-half selects K slots {0,1} vs {2,3}
    const int n  = m;           // B/D column carried by lane

#pragma unroll
    for (int dy = 0; dy < 7; ++dy) {
#pragma unroll
      for (int kc = 0; kc < 6; ++kc) {
        const int jx = kc * 4 + (hi ? 2 : 0);   // even -> 8B aligned pairs
        const v2f a = *(const v2f*)&patch[m + dy][jx];
        const v2f b = *(const v2f*)&Bmat[dy][n][jx];
        acc = __builtin_amdgcn_wmma_f32_16x16x4_f32(
            /*neg_a=*/false, a, /*neg_b=*/false, b,
            /*c_mod=*/(short)0, acc, /*reuse_a=*/false, /*reuse_b=*/false);
      }
    }
  }

  // D layout: VGPR r -> M = r (lanes 0-15) / M = r+8 (lanes 16-31), N = lane&15
  const int n  = lane & 15;
  const int hi = lane >> 4;
#pragma unroll
  for (int r = 0; r < 8; ++r) {
    const int gy = y0 + (hi ? 8 : 0) + r;
    const int gx = x0 + n;
    if (gy < H_OUT && gx < W_OUT)
      out[gy * W_OUT + gx] = excT[gy * W_OUT + gx] + acc[r];
  }
}

// ---------------------------------------------------------------------------
// Stage 3: LIF recurrence over T per pixel + per-(t,b) partial reductions.
// Thread owns one pixel, streams spikes straight to d_out. Block-level
// partial sums for dcmd (spike * dcmd_w) and net_exc (max(lgmd,0)) written to
// workspace -> deterministic final reduce (no float atomics).
// ---------------------------------------------------------------------------
__global__ __launch_bounds__(256) void lif_kernel(
    const float* __restrict__ lgmd, const float* __restrict__ dcmd_w,
    float* __restrict__ out_spikes, float* __restrict__ partials) {
  const int b     = blockIdx.y;
  const int chunk = blockIdx.x;
  const int nch   = gridDim.x;
  const int p     = chunk * 256 + threadIdx.x;
  const bool valid = (p < NPIX);

  const float wp = valid ? dcmd_w[p] : 0.0f;
  float v = 0.0f;

  __shared__ float red0[256];
  __shared__ float red1[256];

  for (int t = 0; t < T_DIM; ++t) {
    const size_t base = ((size_t)t * B_DIM + b) * NPIX;
    const float xin = valid ? lgmd[base + p] : 0.0f;
    v += (xin - v) * 0.5f;                      // TAU = 2, decay_input
    const float s = (v >= 0.5f) ? 1.0f : 0.0f;  // V_TH = 0.5
    v *= (1.0f - s);                            // hard reset to 0
    if (valid) out_spikes[base + p] = s;

    red0[threadIdx.x] = s * wp;
    red1[threadIdx.x] = fmaxf(xin, 0.0f);
    __syncthreads();
#pragma unroll
    for (int off = 128; off > 0; off >>= 1) {
      if (threadIdx.x < off) {
        red0[threadIdx.x] += red0[threadIdx.x + off];
        red1[threadIdx.x] += red1[threadIdx.x + off];
      }
      __syncthreads();
    }
    if (threadIdx.x == 0) {
      const size_t pb = (((size_t)t * B_DIM + b) * nch + chunk) * 2;
      partials[pb + 0] = red0[0];
      partials[pb + 1] = red1[0];
    }
    __syncthreads();
  }
}

__global__ __launch_bounds__(256) void finalize_kernel(
    const float* __restrict__ partials, float* __restrict__ dcmd,
    float* __restrict__ net_exc, int nch) {
  const int i = blockIdx.x * blockDim.x + threadIdx.x;
  if (i < T_DIM * B_DIM) {
    float a = 0.0f, c = 0.0f;
    for (int k = 0; k < nch; ++k) {
      a += partials[((size_t)i * nch + k) * 2 + 0];
      c += partials[((size_t)i * nch + k) * 2 + 1];
    }
    dcmd[i]    = a;                         // dcmd_w already carries 1/(h*w)
    net_exc[i] = c * (1.0f / (float)NPIX);  // spatial mean of clip(lgmd, 0)
  }
}

// ---------------------------------------------------------------------------
extern "C" void kernel_launch(void* const* d_in, const int* in_sizes, int n_in,
                              void* d_out, int out_size, void* d_ws,
                              size_t ws_size, hipStream_t stream) {
  (void)in_sizes; (void)n_in; (void)out_size; (void)ws_size;
  const float* x      = (const float*)d_in[0];
  const float* exc_w  = (const float*)d_in[1];
  const float* inh_w  = (const float*)d_in[2];
  const float* dcmd_w = (const float*)d_in[3];

  float* out     = (float*)d_out;
  float* dcmd    = out;
  float* spikes  = out + T_DIM * B_DIM;
  float* net_exc = out + T_DIM * B_DIM + (size_t)T_DIM * B_DIM * NPIX;

  float* ws       = (float*)d_ws;
  float* exc      = ws;                                  // 8,944,000 floats
  float* lgmd     = ws + (size_t)T_DIM * B_DIM * NPIX;   // 8,944,000 floats
  float* partials = lgmd + (size_t)T_DIM * B_DIM * NPIX; // 70,400 floats

  // Stage 1: pool + exc conv (HBM-bound pass over x)
  dim3 g1((W_OUT + 31) / 32, (H_OUT + 31) / 32, T_DIM * B_DIM);  // 3,3,1600
  pool_exc_kernel<<<g1, 256, 0, stream>>>(x, exc_w, exc);

  // Stage 2: 7x7 inh conv via f32 WMMA Toeplitz GEMM, one wave per 16x16 tile
  dim3 g2((W_OUT + 15) / 16, (H_OUT + 15) / 16, T_DIM * B_DIM);  // 6,5,1600
  lgmd_inh_wmma_kernel<<<g2, 32, 0, stream>>>(exc, inh_w, lgmd);

  // Stage 3: LIF + partial reductions
  const int nch = (NPIX + 255) / 256;  // 22
  dim3 g3(nch, B_DIM);
  lif_kernel<<<g3, 256, 0, stream>>>(lgmd, dcmd_w, spikes, partials);

  // Stage 4: deterministic final reduce for dcmd / net_exc
  finalize_kernel<<<(T_DIM * B_DIM + 255) / 256, 256, 0, stream>>>(
      partials, dcmd, net_exc, nch);
}